// PoCooLoss_33200097198544
// MI455X (gfx1250) — compile-verified
//
#include <hip/hip_runtime.h>
#include <math.h>

typedef __attribute__((ext_vector_type(2))) float v2f;
typedef __attribute__((ext_vector_type(8))) float v8f;

#define J_GT   64
#define C_CLS  80
#define BLOCK  256
#define IOU_THR 0.5f

// ---------------------------------------------------------------------------
// Kernel 1: zero the scalar accumulator in workspace (d_ws is poisoned 0xAA).
// ---------------------------------------------------------------------------
__global__ void pocoo_zero_kernel(float* __restrict__ ws) {
    if (threadIdx.x == 0 && blockIdx.x == 0) ws[0] = 0.0f;
}

// ---------------------------------------------------------------------------
// Kernel 2: fused IoU-argmax + BCE loss, scalar reduction via WMMA + atomics.
// Grid: (ceil(M/BLOCK), B). One thread per (b, m).
// ---------------------------------------------------------------------------
__global__ __launch_bounds__(BLOCK) void pocoo_loss_kernel(
    const float* __restrict__ boxes_pred,   // [B, M, 4] (cx,cy,w,h)
    const float* __restrict__ scores,       // [B, M, C]
    const float* __restrict__ gt_boxes,     // [B, J, 4]
    const float* __restrict__ gt_labels,    // [B, J, C]
    const int*   __restrict__ Hptr,
    const int*   __restrict__ Wptr,
    int M,
    float* __restrict__ gsum)
{
    __shared__ float s_gt[J_GT * 4];         // 1 KB
    __shared__ float s_lab[J_GT * C_CLS];    // 20 KB
    __shared__ float s_part[BLOCK / 32];

    const int b   = blockIdx.y;
    const int tid = threadIdx.x;
    const int m   = blockIdx.x * BLOCK + tid;

    // Stage GT boxes + labels for this batch into LDS (float4, all aligned).
    {
        const float4* src = (const float4*)(gt_boxes + (size_t)b * J_GT * 4);
        float4*       dst = (float4*)s_gt;
        for (int i = tid; i < J_GT; i += BLOCK) dst[i] = src[i];
    }
    {
        const float4* src = (const float4*)(gt_labels + (size_t)b * J_GT * C_CLS);
        float4*       dst = (float4*)s_lab;
        for (int i = tid; i < (J_GT * C_CLS) / 4; i += BLOCK) dst[i] = src[i];
    }
    __syncthreads();

    const float Wf = (float)Wptr[0];
    const float Hf = (float)Hptr[0];

    float acc = 0.0f;
    if (m < M) {
        // ---- IoU argmax against 64 GT boxes (from LDS) ----
        const float4 pb = ((const float4*)(boxes_pred + ((size_t)b * M + m) * 4))[0];
        const float px1 = pb.x - pb.z * 0.5f, px2 = pb.x + pb.z * 0.5f;
        const float py1 = pb.y - pb.w * 0.5f, py2 = pb.y + pb.w * 0.5f;
        const float parea = (px2 - px1) * (py2 - py1);

        float best = -1.0f;
        int   bj   = 0;
        #pragma unroll 4
        for (int j = 0; j < J_GT; ++j) {
            const float4 gb = ((const float4*)s_gt)[j];
            const float gx1 = gb.x - gb.z * 0.5f, gx2 = gb.x + gb.z * 0.5f;
            const float gy1 = gb.y - gb.w * 0.5f, gy2 = gb.y + gb.w * 0.5f;
            float iw = fminf(px2, gx2) - fmaxf(px1, gx1);
            float ih = fminf(py2, gy2) - fmaxf(py1, gy1);
            iw = fmaxf(iw, 0.0f);
            ih = fmaxf(ih, 0.0f);
            const float inter = iw * ih;
            const float garea = (gx2 - gx1) * (gy2 - gy1);
            const float iou   = inter / (parea + garea - inter + 1e-6f);
            if (iou > best) { best = iou; bj = j; }   // first-max, matches argmax
        }

        // ---- BCE over 80 classes ----
        const float4* p4 = (const float4*)(scores + ((size_t)b * M + m) * C_CLS);
        if (best > IOU_THR) {
            // positive: bce(p, y) * (size_factor + 1)
            const float bw = s_gt[bj * 4 + 2];
            const float bh = s_gt[bj * 4 + 3];
            const float sf = sqrtf(1.0f - sqrtf((bw / Wf) * (bh / Hf)));  // ALPHA = 0.5
            const float w  = sf + 1.0f;
            const float4* y4 = (const float4*)(s_lab + bj * C_CLS);
            #pragma unroll 4
            for (int c = 0; c < C_CLS / 4; ++c) {
                const float4 p = p4[c];
                const float4 y = y4[c];
                const float pv[4] = {p.x, p.y, p.z, p.w};
                const float yv[4] = {y.x, y.y, y.z, y.w};
                #pragma unroll
                for (int k = 0; k < 4; ++k) {
                    const float lp  = fmaxf(logf(pv[k]),        -100.0f);
                    const float l1m = fmaxf(logf(1.0f - pv[k]), -100.0f);
                    acc += -(yv[k] * lp + (1.0f - yv[k]) * l1m) * w;
                }
            }
        } else {
            // negative: bce(p, 0) * p^2 = -log(1-p) * p^2
            #pragma unroll 4
            for (int c = 0; c < C_CLS / 4; ++c) {
                const float4 p = p4[c];
                const float pv[4] = {p.x, p.y, p.z, p.w};
                #pragma unroll
                for (int k = 0; k < 4; ++k) {
                    const float l1m = fmaxf(logf(1.0f - pv[k]), -100.0f);
                    acc += -l1m * pv[k] * pv[k];
                }
            }
        }
    }

    // ---- Wave32 reduction via the matrix unit ----
    // A = ones(16x4), B holds the 32 lane partials in its first VGPR (zeros in
    // the second). D[m,n] = sum_k B[k,n]: one v_wmma_f32_16x16x4_f32 collapses
    // 32 partials into 16 column sums (broadcast across rows); 4 xor-shuffles
    // finish the wave total. EXEC is all-1s here (no divergence at this point).
    v2f av; av.x = 1.0f; av.y = 1.0f;
    v2f bv; bv.x = acc;  bv.y = 0.0f;
    v8f cz = {};
    v8f d = __builtin_amdgcn_wmma_f32_16x16x4_f32(
        /*neg_a=*/false, av, /*neg_b=*/false, bv,
        /*c_mod=*/(short)0, cz, /*reuse_a=*/false, /*reuse_b=*/false);
    float r = d[0];
    r += __shfl_xor(r, 1, 32);
    r += __shfl_xor(r, 2, 32);
    r += __shfl_xor(r, 4, 32);
    r += __shfl_xor(r, 8, 32);   // lane 0 now holds the wave total

    const int lane = tid & 31;
    const int wave = tid >> 5;
    if (lane == 0) s_part[wave] = r;
    __syncthreads();
    if (tid == 0) {
        float t = 0.0f;
        #pragma unroll
        for (int i = 0; i < BLOCK / 32; ++i) t += s_part[i];
        atomicAdd(gsum, t);
    }
}

// ---------------------------------------------------------------------------
// Kernel 3: finalize — out[0] = sum / B
// ---------------------------------------------------------------------------
__global__ void pocoo_final_kernel(const float* __restrict__ ws,
                                   float* __restrict__ out, float invB) {
    if (threadIdx.x == 0 && blockIdx.x == 0) out[0] = ws[0] * invB;
}

// ---------------------------------------------------------------------------
extern "C" void kernel_launch(void* const* d_in, const int* in_sizes, int n_in,
                              void* d_out, int out_size, void* d_ws, size_t ws_size,
                              hipStream_t stream) {
    const float* boxes_pred = (const float*)d_in[0];   // B_final           [B,M,4]
    const float* scores     = (const float*)d_in[1];   // final_class_scores[B,M,C]
    const float* gt_boxes   = (const float*)d_in[2];   // gt_boxes          [B,J,4]
    const float* gt_labels  = (const float*)d_in[3];   // gt_labels         [B,J,C]
    const int*   Hptr       = (const int*)d_in[4];     // scalar H
    const int*   Wptr       = (const int*)d_in[5];     // scalar W

    const int B = in_sizes[2] / (J_GT * 4);            // B*J*4 / (J*4)
    const int M = in_sizes[0] / (4 * B);               // B*M*4 / (4*B)

    float* ws  = (float*)d_ws;
    float* out = (float*)d_out;

    pocoo_zero_kernel<<<1, 32, 0, stream>>>(ws);

    dim3 grid((M + BLOCK - 1) / BLOCK, B);
    pocoo_loss_kernel<<<grid, BLOCK, 0, stream>>>(
        boxes_pred, scores, gt_boxes, gt_labels, Hptr, Wptr, M, ws);

    pocoo_final_kernel<<<1, 32, 0, stream>>>(ws, out, 1.0f / (float)B);
}